// AFNO3DPrompt_14259291423321
// MI455X (gfx1250) — compile-verified
//
#include <hip/hip_runtime.h>
#include <hip/hip_bf16.h>

// ---------------------------------------------------------------------------
// AFNO3D prompt block on MI455X (gfx1250), wave32, fp32 WMMA (16x16x4).
//
// Pipeline (all stages are GEMMs on v_wmma_f32_16x16x4_f32):
//   K0 bias_reduce : bias_pre[c], bias_post[c] = sum_p prompt[p,c]
//   K1 init_tables : DFT matrices pre-baked in WMMA A-operand lane layout
//   K2 lfwd        : real DFT along L (32 -> 8 modes), rows 0-7 = Re, 8-15 = Im
//   K3 dft_stage   : complex DFT along H (64 -> 32 modes), K-doubled GEMM,
//                    B tiles staged through LDS by the Tensor Data Mover
//   K4 dft_stage   : complex DFT along W (64 -> 32 modes)
//   K5 mlp         : per-block complex 2-layer MLP with exact GELU (erf)
//   K6 dft_stage   : inverse along W (32 modes -> 64)
//   K7 dft_stage   : inverse along H (32 modes -> 64)
//   K8 linv        : inverse real DFT along L + residual + pre/post bias
// ---------------------------------------------------------------------------

typedef float v2f __attribute__((ext_vector_type(2)));
typedef float v8f __attribute__((ext_vector_type(8)));
typedef unsigned int v4u __attribute__((ext_vector_type(4)));
typedef int v4i __attribute__((ext_vector_type(4)));
typedef int v8i __attribute__((ext_vector_type(8)));

static __device__ __forceinline__ v8f wmma4(v2f a, v2f b, v8f c) {
  // D = A(16x4 f32) * B(4x16 f32) + C(16x16 f32)
  return __builtin_amdgcn_wmma_f32_16x16x4_f32(false, a, false, b, (short)0, c,
                                               false, false);
}

static __device__ __forceinline__ float gelu_exact(float v) {
  return 0.5f * v * (1.0f + erff(v * 0.70710678118654752f));
}

#define TWO_PI 6.28318530717958647692f
#define INV_SQRT32 0.17677669529663688f /* 1/sqrt(32) */

// ---------------------------------------------------------------------------
// Tensor Data Mover: DMA a [rows x 16] fp32 tile (row stride = rowStrideElems
// elements) from global into LDS at ldsByteAddr, packed row-major [rows][16].
// D# built per CDNA5 ISA 8.3/8.4 (group0: count/lds/global/type, group1:
// data_size=4B, tile dims, dim0 stride).  Tracked by TENSORcnt.
// 6-arg builtin form (clang-23 / therock headers): extra zero SGPR group.
// ---------------------------------------------------------------------------
static __device__ __forceinline__ void tdm_load_tile(const float* src,
                                                     unsigned ldsByteAddr,
                                                     int rows,
                                                     long rowStrideElems) {
  unsigned long long ga = (unsigned long long)(uintptr_t)src;
  v4u g0;
  g0.x = 1u;                                   // count=1, user mode, no gather
  g0.y = ldsByteAddr;                          // lds_addr
  g0.z = (unsigned)(ga & 0xffffffffull);       // global_addr[31:0]
  g0.w = (unsigned)((ga >> 32) & 0x01ffffffu)  // global_addr[56:32]
         | (2u << 30);                         // type = 2 ("image")
  v8i g1;
  g1[0] = (int)(2u << 16);                     // data_size = 2 -> 4 bytes
  g1[1] = 0;                                   // tensor_dim0[15:0] (td0=1<<30)
  g1[2] = 0x4000;                              // td0[31:16]; td1[15:0]=0
  g1[3] = (int)(0x4000u | (16u << 16));        // td1[31:16]; tile_dim0 = 16
  g1[4] = rows;                                // tile_dim1 = rows; tile_dim2=0
  g1[5] = (int)(unsigned)rowStrideElems;       // tensor_dim0_stride[31:0]
  g1[6] = 0;                                   // stride[47:32]; td1_stride lo
  g1[7] = 0;
  v4i z4 = {0, 0, 0, 0};
  v8i z8 = {0, 0, 0, 0, 0, 0, 0, 0};
  __builtin_amdgcn_tensor_load_to_lds(g0, g1, z4, z4, z8, 0);
}

// ---------------------------------------------------------------------------
// K0: reduce prompts over prompt_dim -> per-channel biases
// pre/post prompt flat layout: [prompt_dim=32][C=256]
// ---------------------------------------------------------------------------
__global__ void bias_reduce(const float* __restrict__ pre,
                            const float* __restrict__ post,
                            float* __restrict__ bPre, float* __restrict__ bPost) {
  int c = threadIdx.x; // 256 threads
  float s0 = 0.f, s1 = 0.f;
  for (int p = 0; p < 32; ++p) {
    s0 += pre[p * 256 + c];
    s1 += post[p * 256 + c];
  }
  bPre[c] = s0;
  bPost[c] = s1;
}

// ---------------------------------------------------------------------------
// K1: bake DFT matrices into WMMA A-operand layout.
// A-operand layout for 16x16x4 f32 (per ISA): lane<16: M=lane, K={4kc,4kc+1};
// lane>=16: M=lane-16, K={4kc+2,4kc+3}.  Flat: ((mt*KC+kc)*32+lane)*2+e.
// ---------------------------------------------------------------------------
__global__ void init_tables(float* __restrict__ TL, float* __restrict__ THWr,
                            float* __restrict__ THWi, float* __restrict__ TINVr,
                            float* __restrict__ TINVi, float* __restrict__ TLI) {
  // Forward L-DFT (real): rows 0-7 = cos (Re), rows 8-15 = -sin (Im); K = l (32)
  for (int idx = threadIdx.x; idx < 512; idx += 256) {
    int e = idx & 1, lane = (idx >> 1) & 31, kc = idx >> 6;
    int m = lane & 15;
    int k = kc * 4 + ((lane < 16) ? e : 2 + e);
    float v;
    if (m < 8)
      v = cosf(TWO_PI * (float)((m * k) & 31) / 32.0f);
    else
      v = -sinf(TWO_PI * (float)(((m - 8) * k) & 31) / 32.0f);
    TL[idx] = v * INV_SQRT32;
  }
  // Forward H/W DFT: e^{-i 2pi m h/64}/8; K-doubled K=128 (k<64 -> Re(B), else Im(B))
  // A'r = [Ar | -Ai] = [cos | sin],  A'i = [Ai | Ar] = [-sin | cos]
  for (int idx = threadIdx.x; idx < 4096; idx += 256) {
    int e = idx & 1, lane = (idx >> 1) & 31, kc = (idx >> 6) & 31, mt = idx >> 11;
    int m = mt * 16 + (lane & 15);
    int k = kc * 4 + ((lane < 16) ? e : 2 + e);
    int h = k & 63;
    float ang = TWO_PI * (float)((m * h) & 63) / 64.0f;
    float c = cosf(ang) * 0.125f, sn = sinf(ang) * 0.125f;
    THWr[idx] = (k < 64) ? c : sn;
    THWi[idx] = (k < 64) ? -sn : c;
  }
  // Inverse H/W DFT: e^{+i 2pi o m/64}/8; K_c=32 modes -> K=64 doubled
  // Ar=cos, Ai=+sin; A'r=[Ar|-Ai], A'i=[Ai|Ar]
  for (int idx = threadIdx.x; idx < 4096; idx += 256) {
    int e = idx & 1, lane = (idx >> 1) & 31, kc = (idx >> 6) & 15, mt = idx >> 10;
    int o = mt * 16 + (lane & 15);
    int k = kc * 4 + ((lane < 16) ? e : 2 + e);
    int mm = k & 31;
    float ang = TWO_PI * (float)((o * mm) & 63) / 64.0f;
    float c = cosf(ang) * 0.125f, sn = sinf(ang) * 0.125f;
    TINVr[idx] = (k < 32) ? c : -sn;
    TINVi[idx] = (k < 32) ? sn : c;
  }
  // Inverse L (irfft w/ zero-padded t>=8):
  // y_l = (1/sqrt32)(Re X0 + 2 sum_{t=1..7} [Xr cos - Xi sin]); K=16 (8 Re + 8 Im)
  for (int idx = threadIdx.x; idx < 512; idx += 256) {
    int e = idx & 1, lane = (idx >> 1) & 31, kc = (idx >> 6) & 3, mt = idx >> 8;
    int l = mt * 16 + (lane & 15);
    int k = kc * 4 + ((lane < 16) ? e : 2 + e); // 0..15
    float v;
    if (k < 8) {
      int t = k;
      v = (t == 0) ? 1.0f : 2.0f * cosf(TWO_PI * (float)((t * l) & 31) / 32.0f);
    } else {
      int t = k - 8;
      v = (t == 0) ? 0.0f : -2.0f * sinf(TWO_PI * (float)((t * l) & 31) / 32.0f);
    }
    TLI[idx] = v * INV_SQRT32;
  }
}

// ---------------------------------------------------------------------------
// K2: forward real DFT along L.  One block per (b,h,w) slice; 8 waves x 2 c-tiles.
// x slice [32 l x 256 c] -> F1r/F1i slice [8 t x 256 c].  Pre-bias folded in.
// ---------------------------------------------------------------------------
__global__ void lfwd(const float* __restrict__ x, const float* __restrict__ bPre,
                     const float* __restrict__ TL, float* __restrict__ F1r,
                     float* __restrict__ F1i) {
  const int lane = threadIdx.x & 31, wave = threadIdx.x >> 5;
  const int lo = lane & 15;
  const bool hi = lane >= 16;
  const long s = blockIdx.x;        // 8192 slices
  const long xBase = s * 8192;      // L*C
  const long fBase = s * 2048;      // T*C
  for (int ct = wave * 2; ct < wave * 2 + 2; ++ct) {
    const int c = ct * 16 + lo;
    const float bias = bPre[c];
    v8f acc = {};
    for (int kc = 0; kc < 8; ++kc) { // K = 32 (l)
      v2f Av = *(const v2f*)(TL + (kc * 32 + lane) * 2);
      int l0 = kc * 4 + (hi ? 2 : 0);
      v2f Bv;
      Bv.x = x[xBase + (long)l0 * 256 + c] + bias;
      Bv.y = x[xBase + (long)(l0 + 1) * 256 + c] + bias;
      acc = wmma4(Av, Bv, acc);
    }
    // D rows: lanes<16 -> t=0..7 (Re), lanes>=16 -> t=0..7 (Im)
    float* dst = hi ? F1i : F1r;
    for (int r = 0; r < 8; ++r) dst[fBase + (long)r * 256 + c] = acc[r];
  }
}

// ---------------------------------------------------------------------------
// K3/K4/K6/K7: generic truncated complex DFT along one axis, K-doubled.
// grid = B * nOuter * mtiles blocks of 256; each wave covers 16 of 128 n-tiles
// over the contiguous (t,c) = 2048-wide inner dimension.
//
// B operands ([kHalf x 16] fp32 tiles, row stride inSK) are staged into LDS by
// the Tensor Data Mover, double-buffered across n-tile iterations; the wave
// overlaps WMMA issue on buffer p with TDM fill of buffer p^1 and synchronizes
// with s_wait_tensorcnt (per-wave in-order TDM completion).
// ---------------------------------------------------------------------------
__global__ void dft_stage(const float* __restrict__ inR, const float* __restrict__ inI,
                          float* __restrict__ outR, float* __restrict__ outI,
                          const float* __restrict__ tabR, const float* __restrict__ tabI,
                          int nOuter, int mtiles, int kcCount, int kHalf,
                          long inSB, long inSO, long inSK,
                          long outSB, long outSO, long outSR) {
  // [dbuf][wave][plane][row<=64][col16] = 128 KB
  __shared__ float smem[2 * 8 * 2 * 64 * 16];

  int bx = blockIdx.x;
  const int mt = bx % mtiles; bx /= mtiles;
  const int outer = bx % nOuter;
  const int b = bx / nOuter;
  const int lane = threadIdx.x & 31, wave = threadIdx.x >> 5;
  const int lo = lane & 15;
  const bool hi = lane >= 16;
  const long inBase = (long)b * inSB + (long)outer * inSO;
  const long outBase = (long)b * outSB + (long)outer * outSO;
  const float* tR = tabR + (long)mt * kcCount * 64;
  const float* tI = tabI + (long)mt * kcCount * 64;
  const int kMask = kHalf - 1;

  const unsigned ldsBase = (unsigned)(unsigned long long)(uintptr_t)smem;
  // byte offset of [buf][wave][plane] slot (64*16 floats per slot)
  const unsigned slotR0 = (unsigned)(((0 * 8 + wave) * 2 + 0) * 1024) * 4u;
  const unsigned slotStrideBuf = (unsigned)(8 * 2 * 64 * 16) * 4u;
  const unsigned slotStridePl = (unsigned)(64 * 16) * 4u;

  // prologue: fill buffer 0 with the first n-tile for this wave
  tdm_load_tile(inR + inBase + wave * 16, ldsBase + slotR0, kHalf, inSK);
  tdm_load_tile(inI + inBase + wave * 16, ldsBase + slotR0 + slotStridePl, kHalf, inSK);

  int buf = 0;
  for (int nt = wave; nt < 128; nt += 8, buf ^= 1) {
    if (nt + 8 < 128) {
      const unsigned s2 = ldsBase + slotR0 + (unsigned)(buf ^ 1) * slotStrideBuf;
      tdm_load_tile(inR + inBase + (nt + 8) * 16, s2, kHalf, inSK);
      tdm_load_tile(inI + inBase + (nt + 8) * 16, s2 + slotStridePl, kHalf, inSK);
      // current pair + next pair outstanding; <=2 means current tile landed
      asm volatile("s_wait_tensorcnt 0x2" ::: "memory");
    } else {
      asm volatile("s_wait_tensorcnt 0x0" ::: "memory");
    }
    const float* tr = smem + ((buf * 8 + wave) * 2 + 0) * 1024;
    const float* ti = smem + ((buf * 8 + wave) * 2 + 1) * 1024;

    const int n = nt * 16 + lo;
    v8f accR = {}, accI = {};
    for (int kc = 0; kc < kcCount; ++kc) {
      v2f Ar = *(const v2f*)(tR + (kc * 32 + lane) * 2);
      v2f Ai = *(const v2f*)(tI + (kc * 32 + lane) * 2);
      const int k0 = kc * 4 + (hi ? 2 : 0);
      const int k1 = k0 + 1;
      const float* p0 = (k0 < kHalf) ? tr : ti;
      const float* p1 = (k1 < kHalf) ? tr : ti;
      v2f Bv;
      Bv.x = p0[(k0 & kMask) * 16 + lo];
      Bv.y = p1[(k1 & kMask) * 16 + lo];
      accR = wmma4(Ar, Bv, accR);
      accI = wmma4(Ai, Bv, accI);
    }
    const int rowBase = mt * 16 + (hi ? 8 : 0);
    for (int r = 0; r < 8; ++r) {
      const long oa = outBase + (long)(rowBase + r) * outSR + n;
      outR[oa] = accR[r];
      outI[oa] = accI[r];
    }
  }
}

// ---------------------------------------------------------------------------
// K5: per-block complex MLP in Fourier space.
// grid = 1024 mode-tiles (16 modes each) x 256 threads; wave = block index nb.
// A operand = mode rows (K-doubled [Xr|Xi]); B operand = weights gathered per lane.
// GELU on accumulators; LDS round-trip converts D layout -> A layout for layer 2.
// ---------------------------------------------------------------------------
__global__ void mlp(const float* __restrict__ F3r, const float* __restrict__ F3i,
                    const float* __restrict__ w1, const float* __restrict__ b1,
                    const float* __restrict__ w2, const float* __restrict__ b2,
                    float* __restrict__ Gr, float* __restrict__ Gi) {
  __shared__ float lds[8][2][16][32]; // [wave(nb)][re/im][m][i] = 32 KB
  const int lane = threadIdx.x & 31, nb = threadIdx.x >> 5;
  const int lo = lane & 15;
  const bool hi = lane >= 16;
  const long qt = blockIdx.x; // mode tile

  // ---- layer 1: O1 = gelu(X * W1 + b1) ----
  v8f o1r[2] = {}, o1i[2] = {};
  for (int kc = 0; kc < 16; ++kc) { // K = 64 (32 Re + 32 Im)
    const int k0 = kc * 4 + (hi ? 2 : 0);
    const int k1 = k0 + 1;
    const int i0 = k0 & 31, i1 = k1 & 31;
    const long arow = (qt * 16 + lo) * 256 + nb * 32;
    const float* pa0 = (k0 < 32) ? F3r : F3i;
    const float* pa1 = (k1 < 32) ? F3r : F3i;
    v2f Av;
    Av.x = pa0[arow + i0];
    Av.y = pa1[arow + i1];
    for (int ntl = 0; ntl < 2; ++ntl) {
      const int o = ntl * 16 + lo;
      const float wr0 = w1[((0 * 8 + nb) * 32 + i0) * 32 + o];
      const float wi0 = w1[((1 * 8 + nb) * 32 + i0) * 32 + o];
      const float wr1 = w1[((0 * 8 + nb) * 32 + i1) * 32 + o];
      const float wi1 = w1[((1 * 8 + nb) * 32 + i1) * 32 + o];
      v2f Br, Bi;
      Br.x = (k0 < 32) ? wr0 : -wi0;
      Br.y = (k1 < 32) ? wr1 : -wi1;
      Bi.x = (k0 < 32) ? wi0 : wr0;
      Bi.y = (k1 < 32) ? wi1 : wr1;
      o1r[ntl] = wmma4(Av, Br, o1r[ntl]);
      o1i[ntl] = wmma4(Av, Bi, o1i[ntl]);
    }
  }
  // bias + exact gelu, stage to LDS in [m][i] layout
  for (int ntl = 0; ntl < 2; ++ntl) {
    const int o = ntl * 16 + lo;
    const float br = b1[(0 * 8 + nb) * 32 + o];
    const float bi = b1[(1 * 8 + nb) * 32 + o];
    const int mrow = hi ? 8 : 0;
    for (int r = 0; r < 8; ++r) {
      lds[nb][0][mrow + r][o] = gelu_exact(o1r[ntl][r] + br);
      lds[nb][1][mrow + r][o] = gelu_exact(o1i[ntl][r] + bi);
    }
  }
  __syncthreads();

  // ---- layer 2: O2 = O1 * W2 + b2 ----
  v8f o2r[2] = {}, o2i[2] = {};
  for (int kc = 0; kc < 16; ++kc) {
    const int k0 = kc * 4 + (hi ? 2 : 0);
    const int k1 = k0 + 1;
    const int i0 = k0 & 31, i1 = k1 & 31;
    const int pl0 = (k0 < 32) ? 0 : 1, pl1 = (k1 < 32) ? 0 : 1;
    v2f Av;
    Av.x = lds[nb][pl0][lo][i0];
    Av.y = lds[nb][pl1][lo][i1];
    for (int ntl = 0; ntl < 2; ++ntl) {
      const int o = ntl * 16 + lo;
      const float wr0 = w2[((0 * 8 + nb) * 32 + i0) * 32 + o];
      const float wi0 = w2[((1 * 8 + nb) * 32 + i0) * 32 + o];
      const float wr1 = w2[((0 * 8 + nb) * 32 + i1) * 32 + o];
      const float wi1 = w2[((1 * 8 + nb) * 32 + i1) * 32 + o];
      v2f Br, Bi;
      Br.x = (k0 < 32) ? wr0 : -wi0;
      Br.y = (k1 < 32) ? wr1 : -wi1;
      Bi.x = (k0 < 32) ? wi0 : wr0;
      Bi.y = (k1 < 32) ? wi1 : wr1;
      o2r[ntl] = wmma4(Av, Br, o2r[ntl]);
      o2i[ntl] = wmma4(Av, Bi, o2i[ntl]);
    }
  }
  for (int ntl = 0; ntl < 2; ++ntl) {
    const int o = ntl * 16 + lo;
    const float br = b2[(0 * 8 + nb) * 32 + o];
    const float bi = b2[(1 * 8 + nb) * 32 + o];
    const int mrow = hi ? 8 : 0;
    for (int r = 0; r < 8; ++r) {
      const long ga = (qt * 16 + mrow + r) * 256 + nb * 32 + o;
      Gr[ga] = o2r[ntl][r] + br;
      Gi[ga] = o2i[ntl][r] + bi;
    }
  }
}

// ---------------------------------------------------------------------------
// K8: inverse real DFT along L + residual + biases.  One block per (b,h,w).
// ---------------------------------------------------------------------------
__global__ void linv(const float* __restrict__ I2r, const float* __restrict__ I2i,
                     const float* __restrict__ TLI, const float* __restrict__ x,
                     const float* __restrict__ bPre, const float* __restrict__ bPost,
                     float* __restrict__ out) {
  const int lane = threadIdx.x & 31, wave = threadIdx.x >> 5;
  const int lo = lane & 15;
  const bool hi = lane >= 16;
  const long s = blockIdx.x;
  const long fBase = s * 2048;
  const long xBase = s * 8192;
  for (int nt = wave * 2; nt < wave * 2 + 2; ++nt) {
    const int c = nt * 16 + lo;
    const float badd = bPre[c] + bPost[c];
    for (int mt = 0; mt < 2; ++mt) {
      v8f acc = {};
      for (int kc = 0; kc < 4; ++kc) { // K = 16 (8 Re + 8 Im modes)
        v2f Av = *(const v2f*)(TLI + ((mt * 4 + kc) * 32 + lane) * 2);
        const int k0 = kc * 4 + (hi ? 2 : 0);
        const int k1 = k0 + 1;
        const float* p0 = (k0 < 8) ? I2r : I2i;
        const float* p1 = (k1 < 8) ? I2r : I2i;
        v2f Bv;
        Bv.x = p0[fBase + (long)(k0 & 7) * 256 + c];
        Bv.y = p1[fBase + (long)(k1 & 7) * 256 + c];
        acc = wmma4(Av, Bv, acc);
      }
      const int rowBase = mt * 16 + (hi ? 8 : 0);
      for (int r = 0; r < 8; ++r) {
        const long oa = xBase + (long)(rowBase + r) * 256 + c;
        out[oa] = acc[r] + x[oa] + badd;
      }
    }
  }
}

// ---------------------------------------------------------------------------
// Launcher.  Workspace carve (floats):
//   F1 (later I2): 2 x 16,777,216   F2 (later I1): 2 x 8,388,608
//   F3: 2 x 4,194,304               G:  2 x 4,194,304
//   tables + biases: ~18K           total ~268.5 MB
// ---------------------------------------------------------------------------
extern "C" void kernel_launch(void* const* d_in, const int* in_sizes, int n_in,
                              void* d_out, int out_size, void* d_ws, size_t ws_size,
                              hipStream_t stream) {
  (void)in_sizes; (void)n_in; (void)out_size; (void)ws_size;
  const float* x    = (const float*)d_in[0];
  const float* w1   = (const float*)d_in[1];
  const float* b1   = (const float*)d_in[2];
  const float* w2   = (const float*)d_in[3];
  const float* b2   = (const float*)d_in[4];
  const float* pre  = (const float*)d_in[5];
  const float* post = (const float*)d_in[6];
  float* out = (float*)d_out;
  float* ws = (float*)d_ws;

  const size_t O_F1R = 0;
  const size_t O_F1I = O_F1R + 16777216UL;
  const size_t O_F2R = O_F1I + 16777216UL;
  const size_t O_F2I = O_F2R + 8388608UL;
  const size_t O_F3R = O_F2I + 8388608UL;
  const size_t O_F3I = O_F3R + 4194304UL;
  const size_t O_GR  = O_F3I + 4194304UL;
  const size_t O_GI  = O_GR + 4194304UL;
  const size_t O_TL  = O_GI + 4194304UL;
  const size_t O_THWR = O_TL + 512;
  const size_t O_THWI = O_THWR + 4096;
  const size_t O_TINVR = O_THWI + 4096;
  const size_t O_TINVI = O_TINVR + 4096;
  const size_t O_TLI  = O_TINVI + 4096;
  const size_t O_BPRE = O_TLI + 512;
  const size_t O_BPOST = O_BPRE + 256;

  float* F1r = ws + O_F1R; float* F1i = ws + O_F1I;
  float* F2r = ws + O_F2R; float* F2i = ws + O_F2I;
  float* F3r = ws + O_F3R; float* F3i = ws + O_F3I;
  float* Gr  = ws + O_GR;  float* Gi  = ws + O_GI;
  float* I1r = F2r; float* I1i = F2i; // reuse: F2 dead after K4
  float* I2r = F1r; float* I2i = F1i; // reuse: F1 dead after K3
  float* TL = ws + O_TL;
  float* THWr = ws + O_THWR; float* THWi = ws + O_THWI;
  float* TINVr = ws + O_TINVR; float* TINVi = ws + O_TINVI;
  float* TLI = ws + O_TLI;
  float* bPre = ws + O_BPRE; float* bPost = ws + O_BPOST;

  bias_reduce<<<1, 256, 0, stream>>>(pre, post, bPre, bPost);
  init_tables<<<1, 256, 0, stream>>>(TL, THWr, THWi, TINVr, TINVi, TLI);

  // K2: L forward (real -> 8 complex modes)
  lfwd<<<8192, 256, 0, stream>>>(x, bPre, TL, F1r, F1i);

  // K3: H forward. F1[b,h,w,t,c] -> F2[b,m,w,t,c]
  dft_stage<<<2 * 64 * 2, 256, 0, stream>>>(
      F1r, F1i, F2r, F2i, THWr, THWi,
      /*nOuter=*/64, /*mtiles=*/2, /*kc=*/32, /*kHalf=*/64,
      /*inSB=*/8388608L, /*inSO=*/2048L, /*inSK=*/131072L,
      /*outSB=*/4194304L, /*outSO=*/2048L, /*outSR=*/131072L);

  // K4: W forward. F2[b,m,w,t,c] -> F3[b,m,mw,t,c]
  dft_stage<<<2 * 32 * 2, 256, 0, stream>>>(
      F2r, F2i, F3r, F3i, THWr, THWi,
      32, 2, 32, 64,
      4194304L, 131072L, 2048L,
      2097152L, 65536L, 2048L);

  // K5: complex block MLP (16384 modes / 16 per tile)
  mlp<<<1024, 256, 0, stream>>>(F3r, F3i, w1, b1, w2, b2, Gr, Gi);

  // K6: W inverse. G[b,m,mw,t,c] -> I1[b,m,w,t,c]
  dft_stage<<<2 * 32 * 4, 256, 0, stream>>>(
      Gr, Gi, I1r, I1i, TINVr, TINVi,
      32, 4, 16, 32,
      2097152L, 65536L, 2048L,
      4194304L, 131072L, 2048L);

  // K7: H inverse. I1[b,m,w,t,c] -> I2[b,h,w,t,c]
  dft_stage<<<2 * 64 * 4, 256, 0, stream>>>(
      I1r, I1i, I2r, I2i, TINVr, TINVi,
      64, 4, 16, 32,
      4194304L, 2048L, 131072L,
      8388608L, 2048L, 131072L);

  // K8: L inverse + residual + pre/post bias
  linv<<<8192, 256, 0, stream>>>(I2r, I2i, TLI, x, bPre, bPost, out);
}